// SAGE_6296422056697
// MI455X (gfx1250) — compile-verified
//
#include <hip/hip_runtime.h>

typedef float v2f __attribute__((ext_vector_type(2)));
typedef float v8f __attribute__((ext_vector_type(8)));

#define FEAT 128        // K dimension for both layers (IN == HID == 128)
#define ROW_TILES 4     // 16-row tiles per block -> 64 rows/block
#define ROWS_PER_BLOCK (ROW_TILES * 16)

// ---------------------------------------------------------------------------
// deg[dst] += 1 per edge (float so buffer can become inv in place)
// ---------------------------------------------------------------------------
__global__ void sage_deg_kernel(const int* __restrict__ edge_dst,
                                float* __restrict__ deg, int E) {
    int i = blockIdx.x * blockDim.x + threadIdx.x;
    if (i < E) atomicAdd(&deg[edge_dst[i]], 1.0f);
}

// inv[n] = 1 / (deg[n] + 1), in place
__global__ void sage_inv_kernel(float* __restrict__ deg, int N) {
    int i = blockIdx.x * blockDim.x + threadIdx.x;
    if (i < N) deg[i] = 1.0f / (deg[i] + 1.0f);
}

// ---------------------------------------------------------------------------
// One wave32 per edge: lane l moves float4 (features 4l..4l+3) from h[src]
// into agg[dst] with f32 atomics. 512B coalesced read per edge; the 51 MB
// node-feature working set is L2-resident (192 MB L2).
// ---------------------------------------------------------------------------
__global__ void sage_scatter_kernel(const float* __restrict__ h,
                                    const int* __restrict__ edge_src,
                                    const int* __restrict__ edge_dst,
                                    float* __restrict__ agg, int E) {
    int gid  = blockIdx.x * blockDim.x + threadIdx.x;
    int e    = gid >> 5;
    int lane = gid & 31;
    if (e >= E) return;
    int s = edge_src[e];
    int d = edge_dst[e];
    const float4 v = ((const float4*)(h + (size_t)s * FEAT))[lane];
    float* ap = agg + (size_t)d * FEAT + lane * 4;
    atomicAdd(ap + 0, v.x);
    atomicAdd(ap + 1, v.y);
    atomicAdd(ap + 2, v.z);
    atomicAdd(ap + 3, v.w);
}

// ---------------------------------------------------------------------------
// Fused normalize + GEMM + bias (+ReLU):
//   out[64-row block] = act(((agg+self)*inv) @ W + b)
// Block = (Nout/16) waves; wave w owns column tile [w*16, w*16+16).
// Each wave keeps 4 independent accumulators (one per 16-row tile) so the
// v_wmma_f32_16x16x4_f32 chains interleave and hide XDL RAW latency, and each
// B fragment is reused 4x.
// ---------------------------------------------------------------------------
__global__ void sage_gemm_kernel(const float* __restrict__ agg,
                                 const float* __restrict__ hself,
                                 const float* __restrict__ inv,
                                 const float* __restrict__ W,   // [FEAT x Nout] row-major
                                 const float* __restrict__ bias,
                                 float* __restrict__ out,
                                 int N, int Nout, int do_relu) {
    __shared__ float u_s[ROWS_PER_BLOCK][FEAT + 4];   // stride 132: conflict-free

    const int row0 = blockIdx.x * ROWS_PER_BLOCK;

    // Stage normalized 64x128 tile into LDS (all threads cooperate).
    if (row0 + ROWS_PER_BLOCK <= N) {
        for (int idx = threadIdx.x; idx < ROWS_PER_BLOCK * FEAT; idx += blockDim.x) {
            int r = idx >> 7;
            int c = idx & (FEAT - 1);
            size_t g = (size_t)(row0 + r) * FEAT + c;
            u_s[r][c] = (agg[g] + hself[g]) * inv[row0 + r];
        }
    } else {
        for (int idx = threadIdx.x; idx < ROWS_PER_BLOCK * FEAT; idx += blockDim.x) {
            int r = idx >> 7;
            int c = idx & (FEAT - 1);
            int row = row0 + r;
            float u = 0.0f;
            if (row < N) {
                size_t g = (size_t)row * FEAT + c;
                u = (agg[g] + hself[g]) * inv[row];
            }
            u_s[r][c] = u;
        }
    }
    __syncthreads();

    const int wave = threadIdx.x >> 5;
    const int lane = threadIdx.x & 31;
    const int n0   = wave * 16;        // column tile base
    const int half = lane >> 4;        // 0: K pair {0,1}; 1: K pair {2,3}
    const int l15  = lane & 15;

    v8f acc[ROW_TILES];
    #pragma unroll
    for (int t = 0; t < ROW_TILES; ++t) acc[t] = (v8f){};

    for (int kk = 0; kk < FEAT; kk += 4) {
        const int k0 = kk + 2 * half;
        // B 4x16 f32: rows K=k0, K=k0+1 striped over 16 lanes
        const float* wp = W + (size_t)k0 * Nout + n0 + l15;
        v2f b;
        b.x = wp[0];
        b.y = wp[Nout];
        #pragma unroll
        for (int t = 0; t < ROW_TILES; ++t) {
            // A 16x4 f32: lanes 0-15 -> (M=l15, K=k0,k0+1); lanes 16-31 -> K=k0+2,..
            v2f a;
            a.x = u_s[t * 16 + l15][k0];
            a.y = u_s[t * 16 + l15][k0 + 1];
            acc[t] = __builtin_amdgcn_wmma_f32_16x16x4_f32(
                /*neg_a=*/false, a, /*neg_b=*/false, b,
                /*c_mod=*/(short)0, acc[t], /*reuse_a=*/false, /*reuse_b=*/false);
        }
    }

    // C/D 16x16 f32: VGPR i -> M = tile*16 + i + 8*half, N = l15
    const float  bv      = bias[n0 + l15];
    const size_t rstride = (size_t)Nout;
    float* op = out + (size_t)(row0 + 8 * half) * rstride + n0 + l15;

    if (row0 + ROWS_PER_BLOCK <= N) {       // fast path: no per-row guards
        #pragma unroll
        for (int t = 0; t < ROW_TILES; ++t) {
            float* opt = op + (size_t)(t * 16) * rstride;
            #pragma unroll
            for (int i = 0; i < 8; ++i) {
                float v = acc[t][i] + bv;
                if (do_relu) v = fmaxf(v, 0.0f);
                opt[(size_t)i * rstride] = v;
            }
        }
    } else {                                 // tail block only
        #pragma unroll
        for (int t = 0; t < ROW_TILES; ++t) {
            #pragma unroll
            for (int i = 0; i < 8; ++i) {
                int row = row0 + t * 16 + i + 8 * half;
                if (row < N) {
                    float v = acc[t][i] + bv;
                    if (do_relu) v = fmaxf(v, 0.0f);
                    out[(size_t)row * rstride + n0 + l15] = v;
                }
            }
        }
    }
}

// ---------------------------------------------------------------------------
extern "C" void kernel_launch(void* const* d_in, const int* in_sizes, int n_in,
                              void* d_out, int out_size, void* d_ws, size_t ws_size,
                              hipStream_t stream) {
    const float* x    = (const float*)d_in[0];
    const float* W1   = (const float*)d_in[1];
    const float* b1   = (const float*)d_in[2];
    const float* W2   = (const float*)d_in[3];
    const float* b2   = (const float*)d_in[4];
    const int*   esrc = (const int*)d_in[5];
    const int*   edst = (const int*)d_in[6];

    const int N = in_sizes[0] / FEAT;   // 100000
    const int E = in_sizes[5];          // 1600000

    float* ws  = (float*)d_ws;
    float* inv = ws;                                   // N floats (deg -> inv)
    size_t off = ((size_t)N + 63) & ~(size_t)63;       // align next region
    float* agg = ws + off;                             // N*128 floats (reused)
    float* h1  = agg + (size_t)N * FEAT;               // N*128 floats
    float* out = (float*)d_out;                        // N*64 floats

    const int scatter_blocks = (E + 7) / 8;            // 8 edges (waves) / 256-block
    const int row_blocks     = (N + ROWS_PER_BLOCK - 1) / ROWS_PER_BLOCK;

    // degree + inverse normalizer
    hipMemsetAsync(inv, 0, (size_t)N * sizeof(float), stream);
    sage_deg_kernel<<<(E + 255) / 256, 256, 0, stream>>>(edst, inv, E);
    sage_inv_kernel<<<(N + 255) / 256, 256, 0, stream>>>(inv, N);

    // ----- layer 1: agg = scatter(x); h1 = relu(((agg+x)*inv) @ W1 + b1)
    hipMemsetAsync(agg, 0, (size_t)N * FEAT * sizeof(float), stream);
    sage_scatter_kernel<<<scatter_blocks, 256, 0, stream>>>(x, esrc, edst, agg, E);
    sage_gemm_kernel<<<row_blocks, 256, 0, stream>>>(agg, x, inv, W1, b1, h1,
                                                     N, /*Nout=*/128, /*relu=*/1);

    // ----- layer 2: agg = scatter(h1); out = ((agg+h1)*inv) @ W2 + b2
    hipMemsetAsync(agg, 0, (size_t)N * FEAT * sizeof(float), stream);
    sage_scatter_kernel<<<scatter_blocks, 256, 0, stream>>>(h1, esrc, edst, agg, E);
    sage_gemm_kernel<<<row_blocks, 128, 0, stream>>>(agg, h1, inv, W2, b2, out,
                                                     N, /*Nout=*/64, /*relu=*/0);
}